// VectorQuantizer_45526653337911
// MI455X (gfx1250) — compile-verified
//
#include <hip/hip_runtime.h>
#include <stdint.h>

typedef float    v2f  __attribute__((ext_vector_type(2)));
typedef float    v8f  __attribute__((ext_vector_type(8)));
typedef uint32_t u32x4 __attribute__((ext_vector_type(4)));
typedef uint32_t u32x8 __attribute__((ext_vector_type(8)));

#define N_ROWS 65536          // 64*32*32
#define D 64
#define K 1024
#define ESTRIDE 68            // padded LDS row stride (floats): 4-bank skew -> conflict-free B frags
#define ROWS_PER_WAVE 32      // two 16-row C tiles share each B fragment
#define WAVES_PER_BLOCK 8
#define ROWS_PER_BLOCK (ROWS_PER_WAVE * WAVES_PER_BLOCK)   // 256
#define LOSS_OFF (N_ROWS * D)                              // 4194304
#define IDX_OFF (N_ROWS * D + 1)

// LDS: codebook (padded) + row norms + per-row winning index + loss partials
#define SMEM_BYTES ((K * ESTRIDE + K) * sizeof(float) + ROWS_PER_BLOCK * sizeof(int) + WAVES_PER_BLOCK * sizeof(float))

__global__ void vq_init_kernel(float* out) {
    if (threadIdx.x == 0) out[LOSS_OFF] = 0.0f;
}

__device__ __forceinline__ v8f wmma_f32(v2f a, v2f b, v8f c) {
    return __builtin_amdgcn_wmma_f32_16x16x4_f32(false, a, false, b, (short)0, c, false, false);
}

__global__ __launch_bounds__(256) void vq_main_kernel(const float* __restrict__ x,
                                                      const float* __restrict__ emb,
                                                      float* __restrict__ out) {
    extern __shared__ char smem_raw[];
    float* smE   = (float*)smem_raw;                 // [K][ESTRIDE]
    float* smEsq = smE + K * ESTRIDE;                // [K]
    int*   smIdx = (int*)(smEsq + K);                // [ROWS_PER_BLOCK]
    float* smRed = (float*)(smIdx + ROWS_PER_BLOCK); // [WAVES_PER_BLOCK]

    const int tid  = threadIdx.x;
    const int wave = tid >> 5;
    const int lane = tid & 31;

    // ---- TDM: DMA the whole codebook into LDS with 68-dword padded rows ----
    // D# group0: count=1, lds_addr, global_addr[56:0], type=2
    // D# group1: data_size=4B, pad_enable, pad_interval=64 dwords, pad_amount=4 dwords,
    //            tensor_dim0=64, tensor_dim1=1024, tile_dim0=64, tile_dim1=1024, stride0=64
    if (wave == 0) {
        const uint64_t ea  = (uint64_t)emb;
        const uint32_t lds = __builtin_amdgcn_groupstaticsize();   // dynamic-LDS base offset
        u32x4 g0;
        g0[0] = 1u;                                   // count=1, user mode
        g0[1] = lds;                                  // lds_addr (bytes)
        g0[2] = (uint32_t)ea;                         // global_addr[31:0]
        g0[3] = (uint32_t)((ea >> 32) & 0x01FFFFFFu)  // global_addr[56:32]
              | (2u << 30);                           // type=2 ("image")
        u32x8 g1;
        g1[0] = (2u << 16)        // data_size = 4 bytes
              | (1u << 20)        // pad_enable
              | (5u << 22)        // pad_interval: 5 -> every 64 dwords
              | (3u << 25);       // pad_amount:  3 -> 4 dwords (64+4 = ESTRIDE)
        g1[1] = (uint32_t)D << 16;         // tensor_dim0 = 64 (bits 79:48, low part)
        g1[2] = (uint32_t)K << 16;         // tensor_dim1 = 1024 (bits 111:80, low part)
        g1[3] = (uint32_t)D << 16;         // tile_dim0 = 64 (bits 127:112)
        g1[4] = (uint32_t)K;               // tile_dim1 = 1024 (bits 143:128)
        g1[5] = (uint32_t)D;               // tensor_dim0_stride = 64 (bits 207:160, low part)
        g1[6] = 0u;
        g1[7] = 0u;
        asm volatile("tensor_load_to_lds %0, %1" :: "s"(g0), "s"(g1) : "memory");
        __builtin_amdgcn_s_wait_tensorcnt(0);
    }
    __syncthreads();

    // ---- codebook row norms ||e||^2 ----
    for (int r = tid; r < K; r += 256) {
        const float* er = smE + r * ESTRIDE;
        float s = 0.f;
        #pragma unroll
        for (int k2 = 0; k2 < D; ++k2) s += er[k2] * er[k2];
        smEsq[r] = s;
    }
    __syncthreads();

    const int half = lane >> 4;   // ISA f32 16x4 A layout: lanes 16-31 hold K+2,K+3
    const int l16  = lane & 15;
    const int rowBase = blockIdx.x * ROWS_PER_BLOCK + wave * ROWS_PER_WAVE;

    // ---- A fragments for two 16-row tiles: 16 k-steps each, float2 per lane ----
    v2f a0[16], a1[16];
    {
        const float* xr0 = x + (size_t)(rowBase + l16) * D;
        const float* xr1 = xr0 + 16 * D;
        #pragma unroll
        for (int s = 0; s < 16; ++s) {
            const int k0 = 4 * s + 2 * half;
            a0[s].x = xr0[k0]; a0[s].y = xr0[k0 + 1];
            a1[s].x = xr1[k0]; a1[s].y = xr1[k0 + 1];
        }
    }

    // running argmin of (||e||^2 - 2 x.e) per accumulator slot, per tile
    float bd0[8], bd1[8];
    int   bi0[8], bi1[8];
    #pragma unroll
    for (int v = 0; v < 8; ++v) {
        bd0[v] = __builtin_inff(); bi0[v] = 0;
        bd1[v] = __builtin_inff(); bi1[v] = 0;
    }

    for (int c = 0; c < K / 16; ++c) {
        const int n = c * 16 + l16;          // this lane's column (code id)
        const float* erow = smE + n * ESTRIDE;
        v8f acc0 = {}, acc1 = {};
        #pragma unroll
        for (int s = 0; s < 16; ++s) {
            const int k0 = 4 * s + 2 * half;
            v2f b;
            b.x = erow[k0];
            b.y = erow[k0 + 1];
            acc0 = wmma_f32(a0[s], b, acc0);   // one B fragment feeds
            acc1 = wmma_f32(a1[s], b, acc1);   // two independent chains
        }
        const float en = smEsq[n];
        #pragma unroll
        for (int v = 0; v < 8; ++v) {
            const float d0 = en - 2.0f * acc0[v];
            const float d1 = en - 2.0f * acc1[v];
            if (d0 < bd0[v]) { bd0[v] = d0; bi0[v] = n; }   // ascending n => first-min kept
            if (d1 < bd1[v]) { bd1[v] = d1; bi1[v] = n; }
        }
    }

    // ---- argmin reduce across the 16 lanes of each half (C layout: VGPR v = rows v, v+8) ----
    #pragma unroll
    for (int v = 0; v < 8; ++v) {
        float d0 = bd0[v]; int i0 = bi0[v];
        float d1 = bd1[v]; int i1 = bi1[v];
        #pragma unroll
        for (int off = 8; off >= 1; off >>= 1) {
            const float od0 = __shfl_xor(d0, off, 32);
            const int   oi0 = __shfl_xor(i0, off, 32);
            const float od1 = __shfl_xor(d1, off, 32);
            const int   oi1 = __shfl_xor(i1, off, 32);
            if (od0 < d0 || (od0 == d0 && oi0 < i0)) { d0 = od0; i0 = oi0; }
            if (od1 < d1 || (od1 == d1 && oi1 < i1)) { d1 = od1; i1 = oi1; }
        }
        bi0[v] = i0; bi1[v] = i1;
    }
    if (l16 == 0) {
        #pragma unroll
        for (int v = 0; v < 8; ++v) {
            const int m = v + 8 * half;
            smIdx[wave * ROWS_PER_WAVE + m] = bi0[v];
            out[IDX_OFF + rowBase + m] = (float)bi0[v];
            smIdx[wave * ROWS_PER_WAVE + 16 + m] = bi1[v];
            out[IDX_OFF + rowBase + 16 + m] = (float)bi1[v];
        }
    }
    __syncthreads();

    // ---- gather quantized rows from LDS, straight-through output, fused loss ----
    float lsum = 0.f;
    const size_t gbase = (size_t)blockIdx.x * ROWS_PER_BLOCK * D;
    for (int i = tid; i < ROWS_PER_BLOCK * D; i += 256) {
        const int r = i >> 6, col = i & 63;
        const int idx = smIdx[r];
        const float q  = smE[idx * ESTRIDE + col];
        const float xv = x[gbase + i];
        const float diff = q - xv;
        out[gbase + i] = xv + diff;           // x + stopgrad(q - x)
        lsum += diff * diff;
    }
    #pragma unroll
    for (int off = 16; off >= 1; off >>= 1) lsum += __shfl_xor(lsum, off, 32);
    if (lane == 0) smRed[wave] = lsum;
    __syncthreads();
    if (tid == 0) {
        float tot = 0.f;
        #pragma unroll
        for (int w2 = 0; w2 < WAVES_PER_BLOCK; ++w2) tot += smRed[w2];
        // loss = q_latent + 0.25*e_latent = 1.25 * mean((q-x)^2)
        atomicAdd(out + LOSS_OFF, tot * (1.25f / (float)(N_ROWS * D)));
    }
}

extern "C" void kernel_launch(void* const* d_in, const int* in_sizes, int n_in,
                              void* d_out, int out_size, void* d_ws, size_t ws_size,
                              hipStream_t stream) {
    (void)in_sizes; (void)n_in; (void)out_size; (void)d_ws; (void)ws_size;
    const float* x   = (const float*)d_in[0];
    const float* emb = (const float*)d_in[1];
    float* out = (float*)d_out;

    vq_init_kernel<<<1, 1, 0, stream>>>(out);
    vq_main_kernel<<<N_ROWS / ROWS_PER_BLOCK, 256, SMEM_BYTES, stream>>>(x, emb, out);
}